// EdgeODENet_58463094833285
// MI455X (gfx1250) — compile-verified
//
#include <hip/hip_runtime.h>
#include <hip/hip_bf16.h>
#include <math.h>

// ---------------- problem constants ----------------
#define N_NODES 2000
#define N_EDGES 100000
#define F_IN    100000
#define TOT_E   (N_EDGES + N_NODES)   // edges + self loops
#define HALF_E  (N_EDGES / 2)         // 50000
#define DT      0.01f
#define N_STEPS 100

// GEMM tiling: 125 M-tiles of 16 rows, 50 K-splits of 2000
#define MTILES   (N_NODES / 16)       // 125
#define NSPLIT   50
#define KSPLIT   (F_IN / NSPLIT)      // 2000

// ---------------- workspace layout (float offsets) ----------------
#define OFF_HP    0        // 2000*4  pre-attention projection h = hin @ W
#define OFF_HA    8000     // 2000*4  GAT output buffer A
#define OFF_HB    16000    // 2000*4  GAT output buffer B
#define OFF_HS    24000    // 2000    h . a_src
#define OFF_HD    26000    // 2000    h . a_dst
#define OFF_DEN   28000    // 2000    softmax denominator
#define OFF_MAX   30000    // 2000    (uint) softmax max, monotone-mapped
#define OFF_EATT  32000    // 102000  per-edge logit / exp
#define OFF_BIAS6 134000   // 6       bias MLP output for node 0

typedef __attribute__((ext_vector_type(2))) float v2f;
typedef __attribute__((ext_vector_type(8))) float v8f;

__device__ __forceinline__ float lrelu(float v, float slope) {
  return v > 0.f ? v : slope * v;
}
// monotone float<->uint mapping so atomicMax(uint) == float max
__device__ __forceinline__ unsigned fmap(float f) {
  unsigned u = __float_as_uint(f);
  return (u & 0x80000000u) ? ~u : (u | 0x80000000u);
}
__device__ __forceinline__ float funmap(unsigned u) {
  return (u & 0x80000000u) ? __uint_as_float(u & 0x7fffffffu) : __uint_as_float(~u);
}

// ---------------- generic zero ----------------
__global__ void zero_f32(float* p, int n) {
  int i = blockIdx.x * blockDim.x + threadIdx.x;
  if (i < n) p[i] = 0.f;
}

// ---------------- big GEMM: hp[2000][4] = x[2000][100000] @ W[100000][4] ----
// One wave per block. V_WMMA_F32_16X16X4_F32, split-K with atomic reduction.
// A (16x4 f32) layout: lane<16 -> M=lane, K={k,k+1}; lane>=16 -> M=lane-16, K={k+2,k+3}
// B (4x16 f32):        lane<16 -> K={k,k+1} col=lane; lane>=16 -> K={k+2,k+3} col=lane-16
// B columns 4..15 are don't-care (we never read C cols >= 4), so lanes 4..15
// load a clamped column instead of zero-filling -> no exec-mask divergence.
__global__ __launch_bounds__(32) void gemm_xw_wmma(
    const float* __restrict__ x, const float* __restrict__ W,
    float* __restrict__ hp) {
  const int mtile = blockIdx.x;
  const int ks    = blockIdx.y;
  const int lane  = threadIdx.x;
  const int half  = lane >> 4;      // 0/1
  const int n     = lane & 15;      // B/C column (only 0..3 meaningful)
  const int nb    = n & 3;          // clamped B column for don't-care lanes
  const long row  = (long)mtile * 16 + n;
  const long k0   = (long)ks * KSPLIT;

  const float* ax = x + row * (long)F_IN + k0 + half * 2;  // +4 floats per WMMA
  const float* bw = W + (k0 + half * 2) * 4 + nb;          // +16 floats per WMMA

  v8f c = {};
  // outer iter = one 128B cacheline of x per lane-row (32 K = 8 WMMAs);
  // exactly one prefetch per cacheline, no per-WMMA branching.
  for (int kb = 0; kb < KSPLIT; kb += 32) {
    __builtin_prefetch((const void*)(ax + 1024), 0, 1);  // global_prefetch_b8
#pragma unroll
    for (int u = 0; u < 8; ++u) {
      v2f a = *(const v2f*)(ax + u * 4);
      v2f b;
      b.x = bw[u * 16];
      b.y = bw[u * 16 + 4];
      c = __builtin_amdgcn_wmma_f32_16x16x4_f32(false, a, false, b, (short)0, c,
                                                false, false);
    }
    ax += 32;
    bw += 128;
  }
  // C layout: vgpr i -> M = i + (lane>=16 ? 8 : 0), col = lane&15
  if (n < 4) {
    int mbase = mtile * 16 + half * 8;
#pragma unroll
    for (int i = 0; i < 8; ++i)
      atomicAdd(&hp[(mbase + i) * 4 + n], c[i]);
  }
}

// ---------------- GAT attention prep (after projection is in hp) ----------
__global__ void gat_prep(const float* __restrict__ hp,
                         const float* __restrict__ a_src,
                         const float* __restrict__ a_dst,
                         const float* __restrict__ bias,
                         float* hs, float* hd, unsigned* maxb, float* den,
                         float* hout) {
  int nn = blockIdx.x * blockDim.x + threadIdx.x;
  if (nn >= N_NODES) return;
  float h0 = hp[nn * 4 + 0], h1 = hp[nn * 4 + 1];
  float h2 = hp[nn * 4 + 2], h3 = hp[nn * 4 + 3];
  hs[nn] = h0 * a_src[0] + h1 * a_src[1] + h2 * a_src[2] + h3 * a_src[3];
  hd[nn] = h0 * a_dst[0] + h1 * a_dst[1] + h2 * a_dst[2] + h3 * a_dst[3];
  maxb[nn] = 0u;   // below fmap() of any finite float
  den[nn]  = 0.f;
  hout[nn * 4 + 0] = bias[0];
  hout[nn * 4 + 1] = bias[1];
  hout[nn * 4 + 2] = bias[2];
  hout[nn * 4 + 3] = bias[3];
}

// small 4x4 projection fused with prep (layers 2 & 3)
__global__ void gat_project_prep(const float* __restrict__ hin,
                                 const float* __restrict__ W,
                                 const float* __restrict__ a_src,
                                 const float* __restrict__ a_dst,
                                 const float* __restrict__ bias,
                                 float* hp, float* hs, float* hd,
                                 unsigned* maxb, float* den, float* hout) {
  int nn = blockIdx.x * blockDim.x + threadIdx.x;
  if (nn >= N_NODES) return;
  float i0 = hin[nn * 4 + 0], i1 = hin[nn * 4 + 1];
  float i2 = hin[nn * 4 + 2], i3 = hin[nn * 4 + 3];
  float h[4];
#pragma unroll
  for (int j = 0; j < 4; ++j)
    h[j] = i0 * W[j] + i1 * W[4 + j] + i2 * W[8 + j] + i3 * W[12 + j];
  hp[nn * 4 + 0] = h[0]; hp[nn * 4 + 1] = h[1];
  hp[nn * 4 + 2] = h[2]; hp[nn * 4 + 3] = h[3];
  hs[nn] = h[0] * a_src[0] + h[1] * a_src[1] + h[2] * a_src[2] + h[3] * a_src[3];
  hd[nn] = h[0] * a_dst[0] + h[1] * a_dst[1] + h[2] * a_dst[2] + h[3] * a_dst[3];
  maxb[nn] = 0u;
  den[nn]  = 0.f;
  hout[nn * 4 + 0] = bias[0];
  hout[nn * 4 + 1] = bias[1];
  hout[nn * 4 + 2] = bias[2];
  hout[nn * 4 + 3] = bias[3];
}

__device__ __forceinline__ void edge_sd(const int* ei, int e, int& s, int& d) {
  if (e < N_EDGES) { s = ei[e]; d = ei[N_EDGES + e]; }
  else             { s = d = e - N_EDGES; }           // self loop
}

// pass 1: logits + segment max
__global__ void gat_pass1(const int* __restrict__ ei,
                          const float* __restrict__ hs,
                          const float* __restrict__ hd,
                          float* eatt, unsigned* maxb) {
  int e = blockIdx.x * blockDim.x + threadIdx.x;
  if (e >= TOT_E) return;
  int s, d; edge_sd(ei, e, s, d);
  float v = lrelu(hs[s] + hd[d], 0.2f);
  eatt[e] = v;
  atomicMax(&maxb[d], fmap(v));
}

// pass 2: exp + segment sum
__global__ void gat_pass2(const int* __restrict__ ei, float* eatt,
                          const unsigned* __restrict__ maxb, float* den) {
  int e = blockIdx.x * blockDim.x + threadIdx.x;
  if (e >= TOT_E) return;
  int s, d; edge_sd(ei, e, s, d);
  float ex = expf(eatt[e] - funmap(maxb[d]));
  eatt[e] = ex;
  atomicAdd(&den[d], ex);
}

// pass 3: weighted aggregation
__global__ void gat_pass3(const int* __restrict__ ei,
                          const float* __restrict__ eatt,
                          const float* __restrict__ den,
                          const float* __restrict__ hp, float* hout) {
  int e = blockIdx.x * blockDim.x + threadIdx.x;
  if (e >= TOT_E) return;
  int s, d; edge_sd(ei, e, s, d);
  float w = eatt[e] / den[d];
  atomicAdd(&hout[d * 4 + 0], w * hp[s * 4 + 0]);
  atomicAdd(&hout[d * 4 + 1], w * hp[s * 4 + 1]);
  atomicAdd(&hout[d * 4 + 2], w * hp[s * 4 + 2]);
  atomicAdd(&hout[d * 4 + 3], w * hp[s * 4 + 3]);
}

// ---------------- bias MLP, node 0 only (reference uses only bias[0]) -----
__global__ void bias_mlp_kernel(const float* __restrict__ h3,
                                const float* W0, const float* b0,
                                const float* W1, const float* b1,
                                const float* W2, const float* b2,
                                const float* W3, const float* b3,
                                float* out6) {
  if (threadIdx.x != 0 || blockIdx.x != 0) return;
  float in[4] = {h3[0], h3[1], h3[2], h3[3]};
  float t0[10], t1[10], t2[10];
  for (int j = 0; j < 10; ++j) {
    float a = b0[j];
    for (int i = 0; i < 4; ++i) a += in[i] * W0[i * 10 + j];
    t0[j] = lrelu(a, 0.01f);
  }
  for (int j = 0; j < 10; ++j) {
    float a = b1[j];
    for (int i = 0; i < 10; ++i) a += t0[i] * W1[i * 10 + j];
    t1[j] = lrelu(a, 0.01f);
  }
  for (int j = 0; j < 10; ++j) {
    float a = b2[j];
    for (int i = 0; i < 10; ++i) a += t1[i] * W2[i * 10 + j];
    t2[j] = lrelu(a, 0.01f);
  }
  for (int j = 0; j < 6; ++j) {
    float a = b3[j];
    for (int i = 0; i < 10; ++i) a += t2[i] * W3[i * 6 + j];
    out6[j] = a;
  }
}

// ---------------- edge MLP: 9 -> 10 -> 10 -> 10 -> 4, weights in LDS ------
__global__ __launch_bounds__(256) void edge_mlp_kernel(
    const float* __restrict__ h3, const float* __restrict__ ea,
    const int* __restrict__ ei,
    const float* W0, const float* b0, const float* W1, const float* b1,
    const float* W2, const float* b2, const float* W3, const float* b3,
    float* __restrict__ out) {
  __shared__ float s[364];
  int t = threadIdx.x;
  for (int i = t; i < 90; i += 256)  s[i]        = W0[i];
  for (int i = t; i < 10; i += 256)  s[90 + i]   = b0[i];
  for (int i = t; i < 100; i += 256) s[100 + i]  = W1[i];
  for (int i = t; i < 10; i += 256)  s[200 + i]  = b1[i];
  for (int i = t; i < 100; i += 256) s[210 + i]  = W2[i];
  for (int i = t; i < 10; i += 256)  s[310 + i]  = b2[i];
  for (int i = t; i < 40; i += 256)  s[320 + i]  = W3[i];
  for (int i = t; i < 4; i += 256)   s[360 + i]  = b3[i];
  __syncthreads();

  int e = blockIdx.x * 256 + t;
  if (e >= N_EDGES) return;
  int r = ei[e], c = ei[N_EDGES + e];
  float ef[9];
#pragma unroll
  for (int i = 0; i < 4; ++i) ef[i]     = h3[r * 4 + i];
#pragma unroll
  for (int i = 0; i < 4; ++i) ef[4 + i] = h3[c * 4 + i];
  ef[8] = ea[e];

  float t0[10], t1[10], t2[10];
#pragma unroll
  for (int j = 0; j < 10; ++j) {
    float a = s[90 + j];
#pragma unroll
    for (int i = 0; i < 9; ++i) a += ef[i] * s[i * 10 + j];
    t0[j] = lrelu(a, 0.01f);
  }
#pragma unroll
  for (int j = 0; j < 10; ++j) {
    float a = s[200 + j];
#pragma unroll
    for (int i = 0; i < 10; ++i) a += t0[i] * s[100 + i * 10 + j];
    t1[j] = lrelu(a, 0.01f);
  }
#pragma unroll
  for (int j = 0; j < 10; ++j) {
    float a = s[310 + j];
#pragma unroll
    for (int i = 0; i < 10; ++i) a += t1[i] * s[210 + i * 10 + j];
    t2[j] = lrelu(a, 0.01f);
  }
#pragma unroll
  for (int j = 0; j < 4; ++j) {
    float a = s[360 + j];
#pragma unroll
    for (int i = 0; i < 10; ++i) a += t2[i] * s[320 + i * 4 + j];
    out[e * 4 + j] = a;
  }
}

// ---------------- ODE: 100 steps of chained 2x2 matvecs, all in regs ------
__global__ void ode_kernel(const float* __restrict__ x,
                           const float* __restrict__ adj,
                           const float* __restrict__ bias6,
                           float* __restrict__ sout) {
  int n = blockIdx.x * blockDim.x + threadIdx.x;
  if (n >= HALF_E) return;
  float s0 = x[2 * n], s1 = x[2 * n + 1];        // x row 0
  const float* wi = adj + 4 * (long)n;
  const float* wo = adj + 4 * (long)(HALF_E + n);
  float a = wi[0], b = wi[1], c = wi[2], d = wi[3];
  float e = wo[0], f = wo[1], g = wo[2], hh = wo[3];
  float bi0 = bias6[0], bi1 = bias6[1];
  float bo0 = bias6[2], bo1 = bias6[3];
  float g0  = bias6[4], g1  = bias6[5];
#pragma unroll 4
  for (int it = 0; it < N_STEPS; ++it) {
    float o0 = a * s0 + b * s1 + bi0;
    float o1 = c * s0 + d * s1 + bi1;
    float p0 = e * o0 + f * o1 + bo0 + g0 * s0;
    float p1 = g * o0 + hh * o1 + bo1 + g1 * s1;
    s0 += DT * p0;
    s1 += DT * p1;
  }
  sout[2 * n]     = s0;
  sout[2 * n + 1] = s1;
}

// ---------------- host ----------------
struct Idx {
  int bmW[4], bmB[4];   // bias_mlp layer weights / biases
  int emW[4], emB[4];   // edge_mlp
  int gW[3], gAs[3], gAd[3], gB[3];
  int ei;
};

extern "C" void kernel_launch(void* const* d_in, const int* in_sizes, int n_in,
                              void* d_out, int out_size, void* d_ws, size_t ws_size,
                              hipStream_t stream) {
  (void)n_in; (void)out_size; (void)ws_size;
  Idx ix;
  if (in_sizes[2] == 400000) {
    // insertion-order pytree flatten: gat1..3 {W,a_src,a_dst,b}, edge_mlp, bias_mlp
    for (int l = 0; l < 3; ++l) {
      ix.gW[l] = 2 + 4 * l; ix.gAs[l] = 3 + 4 * l;
      ix.gAd[l] = 4 + 4 * l; ix.gB[l] = 5 + 4 * l;
    }
    for (int l = 0; l < 4; ++l) { ix.emW[l] = 14 + 2 * l; ix.emB[l] = 15 + 2 * l; }
    for (int l = 0; l < 4; ++l) { ix.bmW[l] = 22 + 2 * l; ix.bmB[l] = 23 + 2 * l; }
    ix.ei = 30;
  } else {
    // jax sorted-key flatten: bias_mlp, edge_mlp, gat1..3 {W,a_dst,a_src,b}
    for (int l = 0; l < 4; ++l) { ix.bmW[l] = 2 + 2 * l;  ix.bmB[l] = 3 + 2 * l; }
    for (int l = 0; l < 4; ++l) { ix.emW[l] = 10 + 2 * l; ix.emB[l] = 11 + 2 * l; }
    for (int l = 0; l < 3; ++l) {
      ix.gW[l] = 18 + 4 * l; ix.gAd[l] = 19 + 4 * l;
      ix.gAs[l] = 20 + 4 * l; ix.gB[l] = 21 + 4 * l;
    }
    ix.ei = 30;
  }

  const float* x  = (const float*)d_in[0];
  const float* ea = (const float*)d_in[1];
  const int*   ei = (const int*)d_in[ix.ei];
  auto P = [&](int i) { return (const float*)d_in[i]; };

  float*    ws    = (float*)d_ws;
  float*    hp    = ws + OFF_HP;
  float*    hA    = ws + OFF_HA;
  float*    hB    = ws + OFF_HB;
  float*    hs    = ws + OFF_HS;
  float*    hd    = ws + OFF_HD;
  float*    den   = ws + OFF_DEN;
  unsigned* maxb  = (unsigned*)(ws + OFF_MAX);
  float*    eatt  = ws + OFF_EATT;
  float*    bias6 = ws + OFF_BIAS6;
  float*    adj   = (float*)d_out;                 // first 400000 outputs
  float*    sout  = (float*)d_out + 4 * N_EDGES;   // last 100000 outputs

  const int nodeBlocks = (N_NODES + 255) / 256;
  const int edgeBlocks = (TOT_E + 255) / 256;

  // ---- GAT layer 1 (big projection via WMMA, split-K atomics) ----
  zero_f32<<<(N_NODES * 4 + 255) / 256, 256, 0, stream>>>(hp, N_NODES * 4);
  gemm_xw_wmma<<<dim3(MTILES, NSPLIT), 32, 0, stream>>>(x, P(ix.gW[0]), hp);
  gat_prep<<<nodeBlocks, 256, 0, stream>>>(hp, P(ix.gAs[0]), P(ix.gAd[0]),
                                           P(ix.gB[0]), hs, hd, maxb, den, hA);
  gat_pass1<<<edgeBlocks, 256, 0, stream>>>(ei, hs, hd, eatt, maxb);
  gat_pass2<<<edgeBlocks, 256, 0, stream>>>(ei, eatt, maxb, den);
  gat_pass3<<<edgeBlocks, 256, 0, stream>>>(ei, eatt, den, hp, hA);

  // ---- GAT layer 2 ----
  gat_project_prep<<<nodeBlocks, 256, 0, stream>>>(
      hA, P(ix.gW[1]), P(ix.gAs[1]), P(ix.gAd[1]), P(ix.gB[1]),
      hp, hs, hd, maxb, den, hB);
  gat_pass1<<<edgeBlocks, 256, 0, stream>>>(ei, hs, hd, eatt, maxb);
  gat_pass2<<<edgeBlocks, 256, 0, stream>>>(ei, eatt, maxb, den);
  gat_pass3<<<edgeBlocks, 256, 0, stream>>>(ei, eatt, den, hp, hB);

  // ---- GAT layer 3 ----
  gat_project_prep<<<nodeBlocks, 256, 0, stream>>>(
      hB, P(ix.gW[2]), P(ix.gAs[2]), P(ix.gAd[2]), P(ix.gB[2]),
      hp, hs, hd, maxb, den, hA);
  gat_pass1<<<edgeBlocks, 256, 0, stream>>>(ei, hs, hd, eatt, maxb);
  gat_pass2<<<edgeBlocks, 256, 0, stream>>>(ei, eatt, maxb, den);
  gat_pass3<<<edgeBlocks, 256, 0, stream>>>(ei, eatt, den, hp, hA);

  // ---- bias MLP (node 0 only) ----
  bias_mlp_kernel<<<1, 32, 0, stream>>>(hA, P(ix.bmW[0]), P(ix.bmB[0]),
                                        P(ix.bmW[1]), P(ix.bmB[1]),
                                        P(ix.bmW[2]), P(ix.bmB[2]),
                                        P(ix.bmW[3]), P(ix.bmB[3]), bias6);

  // ---- edge MLP -> adj (written straight into d_out) ----
  edge_mlp_kernel<<<(N_EDGES + 255) / 256, 256, 0, stream>>>(
      hA, ea, ei, P(ix.emW[0]), P(ix.emB[0]), P(ix.emW[1]), P(ix.emB[1]),
      P(ix.emW[2]), P(ix.emB[2]), P(ix.emW[3]), P(ix.emB[3]), adj);

  // ---- ODE integration -> s (written into d_out tail) ----
  ode_kernel<<<(HALF_E + 255) / 256, 256, 0, stream>>>(x, adj, bias6, sout);
}